// IzhikevichLayerTorch_57380763075009
// MI455X (gfx1250) — compile-verified
//
#include <hip/hip_runtime.h>
#include <cstdint>

// ---------------------------------------------------------------------------
// Izhikevich layer scan, MI455X (gfx1250, wave32).
//   I:      (T=256, B=64, H=4096) fp32   -> 268 MB streamed in
//   spikes: (T, B, H) fp32 (0.0 / 1.0)   -> 268 MB streamed out
// Pure elementwise recurrent scan: bandwidth-bound (~23 us floor @ 23.3 TB/s).
// Strategy: state (v,u) in registers, 1 thread = 4 neurons, b128 traffic,
// L2 prefetch stream + ASYNCcnt-tracked global->LDS ring for latency hiding,
// non-temporal b128 stores.
// ---------------------------------------------------------------------------

typedef float v4f    __attribute__((ext_vector_type(4)));
typedef int   b128_t __attribute__((vector_size(16)));   // matches builtin param

#define T_STEPS   256
#define N_NEU     (64 * 4096)          // 262144 neurons per timestep
#define VEC       4                    // neurons per thread (float4)
#define THREADS   128                  // 4 wave32 per block
#define BLK_ELEMS (THREADS * VEC)      // 512 floats = 2 KB per block per step
#define NBLOCKS   (N_NEU / BLK_ELEMS)  // 512 blocks
#define DEPTH     8                    // LDS ring slots (power of 2); DEPTH-1 in flight
#define PF_DIST   16                   // L2 prefetch distance in timesteps (16 MB ahead)

// Izhikevich constants (a=0.2, b=2, c=-56, d=-13, dt=1, v_th=30) folded:
//   v_new = 0.04*v^2 + 6*v + 140 - u + I
//   u_new = 0.8*u + 0.4*v
//   spike = v_new > 30 ; then v=-56, u=u_new-13

#if defined(__gfx1250__) &&                                            \
    __has_builtin(__builtin_amdgcn_global_load_async_to_lds_b128) &&   \
    __has_builtin(__builtin_amdgcn_s_wait_asynccnt)
#define USE_ASYNC_LDS 1
#else
#define USE_ASYNC_LDS 0
#endif

// Exact-type address-space casts for the async builtin:
//   arg0: global (AS1) pointer to int __vector_size__(16)
//   arg1: LDS    (AS3) pointer to int __vector_size__(16)
// Integer route always compiles; AS3 ptrs are 32-bit LDS offsets = low 32
// bits of the generic address per the flat->LDS truncation rule.
#define AS1_B128(p) ((__attribute__((address_space(1))) b128_t*)(uintptr_t)(p))
#define AS3_B128(p) ((__attribute__((address_space(3))) b128_t*)(unsigned)(uintptr_t)(p))

__device__ __forceinline__ void izh_step(const v4f& x, float v[VEC], float u[VEC], v4f& s)
{
#pragma unroll
    for (int j = 0; j < VEC; ++j) {
        float vv = v[j], uu = u[j];
        float vn = __builtin_fmaf(vv, __builtin_fmaf(0.04f, vv, 6.0f),
                                  (140.0f - uu) + x[j]);
        float un = __builtin_fmaf(0.8f, uu, 0.4f * vv);
        bool  m  = vn > 30.0f;
        s[j] = m ? 1.0f : 0.0f;
        v[j] = m ? -56.0f : vn;
        u[j] = m ? (un - 13.0f) : un;
    }
}

__global__ __launch_bounds__(THREADS)
void izh_scan_kernel(const float* __restrict__ I, float* __restrict__ S)
{
    const int tid  = threadIdx.x;
    const int base = blockIdx.x * BLK_ELEMS + tid * VEC; // this lane's first neuron

    float v[VEC], u[VEC];
#pragma unroll
    for (int j = 0; j < VEC; ++j) { v[j] = -56.0f; u[j] = -112.0f; }

#if USE_ASYNC_LDS
    __shared__ __align__(16) float lds[DEPTH * BLK_ELEMS];   // 16 KB ring
    const int lbase = tid * VEC;                             // float offset in a slot

    // Prologue: put DEPTH-1 timestep tiles in flight (ASYNCcnt-tracked).
#pragma unroll
    for (int k = 0; k < DEPTH - 1; ++k) {
        __builtin_amdgcn_global_load_async_to_lds_b128(
            AS1_B128(I + (size_t)k * N_NEU + base),
            AS3_B128(&lds[k * BLK_ELEMS + lbase]),
            0, 0);
    }

    for (int t = 0; t < T_STEPS; ++t) {
        // Fire-and-forget HBM->L2 prefetch stream, 16 MB ahead.
        if (t + PF_DIST < T_STEPS)
            __builtin_prefetch(I + (size_t)(t + PF_DIST) * N_NEU + base, 0, 3);

        // Oldest in-flight async load (timestep t, slot t%DEPTH) has retired
        // once ASYNCcnt <= DEPTH-2 (async loads complete in order).
        __builtin_amdgcn_s_wait_asynccnt(DEPTH - 2);
        __asm__ __volatile__("" ::: "memory");

        const int rslot = t & (DEPTH - 1);
        v4f x = *(const v4f*)(&lds[rslot * BLK_ELEMS + lbase]); // ds_load_b128

        v4f s;
        izh_step(x, v, u, s);

        __builtin_nontemporal_store(s, (v4f*)(S + (size_t)t * N_NEU + base));

        // Refill the slot consumed LAST iteration with timestep t+DEPTH-1.
        // (Its ds_load data was already consumed -> no WAR on the ring slot.)
        int tt = t + DEPTH - 1;
        if (tt >= T_STEPS) tt = T_STEPS - 1;   // clamped dummy refill at the tail
        const int wslot = tt & (DEPTH - 1);
        __builtin_amdgcn_global_load_async_to_lds_b128(
            AS1_B128(I + (size_t)tt * N_NEU + base),
            AS3_B128(&lds[wslot * BLK_ELEMS + lbase]),
            0, 0);
    }
    __builtin_amdgcn_s_wait_asynccnt(0);
#else
    // Fallback: register double-buffered NT b128 pipeline + L2 prefetch.
    const v4f* Ip    = (const v4f*)(I + base);
    v4f*       Sp    = (v4f*)(S + base);
    const int  strd  = N_NEU / VEC;

    v4f cur = __builtin_nontemporal_load(Ip);
    for (int t = 0; t < T_STEPS; ++t) {
        if (t + PF_DIST < T_STEPS)
            __builtin_prefetch(I + (size_t)(t + PF_DIST) * N_NEU + base, 0, 3);

        v4f nxt = cur;
        if (t + 1 < T_STEPS)
            nxt = __builtin_nontemporal_load(Ip + (size_t)(t + 1) * strd);

        v4f s;
        izh_step(cur, v, u, s);

        __builtin_nontemporal_store(s, Sp + (size_t)t * strd);
        cur = nxt;
    }
#endif
}

extern "C" void kernel_launch(void* const* d_in, const int* in_sizes, int n_in,
                              void* d_out, int out_size, void* d_ws, size_t ws_size,
                              hipStream_t stream)
{
    (void)in_sizes; (void)n_in; (void)out_size; (void)d_ws; (void)ws_size;
    const float* I = (const float*)d_in[0];
    float*       S = (float*)d_out;
    izh_scan_kernel<<<NBLOCKS, THREADS, 0, stream>>>(I, S);
}